// SNN_33406255628713
// MI455X (gfx1250) — compile-verified
//
#include <hip/hip_runtime.h>

typedef float v2f __attribute__((ext_vector_type(2)));
typedef float v4f __attribute__((ext_vector_type(4)));
typedef float v8f __attribute__((ext_vector_type(8)));

__global__ __launch_bounds__(256) void snn_lif_wmma_kernel(
    const float* __restrict__ x,  const float* __restrict__ W1,
    const float* __restrict__ b1, const float* __restrict__ W2,
    const float* __restrict__ b2, float* __restrict__ out)
{
    constexpr int  T  = 100;
    constexpr long B  = 65536;
    constexpr int  NH = 10, NO = 4;
    constexpr long o_mem1 = (long)T * B * NH;           // after spk1_rec
    constexpr long o_spk2 = 2 * o_mem1;                 // after mem1_rec
    constexpr long o_mem2 = o_spk2 + (long)T * B * NO;  // after spk2_rec

    const int  lane  = threadIdx.x & 31;
    const int  m     = lane & 15;         // A row (lanes 0-15) / batch column N
    const bool lo    = lane < 16;
    const int  hbase = lo ? 0 : 8;        // C layout: high lanes hold M = r+8

    const long tile = (long)blockIdx.x * (blockDim.x >> 5) + (threadIdx.x >> 5);
    const long b0   = tile * 16;          // 16 batch elements per wave

    // ---- Layer-1 A operand (16x4). Rows 0..9 = W1; rows 10..15 DUPLICATE
    // W1 rows 2..7 so the high lane-half carries valid h2..h7 trajectories.
    // Lanes 0-15 supply K=0,1 ; lanes 16-31 supply K=2,3 (zero pad).
    const int mr = (m < NH) ? m : m - 8;
    v2f A1;
    A1.x = lo ? W1[mr * 2 + 0] : 0.0f;
    A1.y = lo ? W1[mr * 2 + 1] : 0.0f;

    // ---- Layer-2 A operand: W2 (4x10) in three 16x4 K-chunks.
    // Rows 0..3 = W2 rows 0..3 ; rows 8..11 DUPLICATE W2 rows 0..3 (so high
    // lanes carry duplicate mem2 trajectories). Chunks 0,1: K=0..3 -> h=4c+K.
    // Chunk 2: K=2,3 -> h8,h9 (weights only on the K=2,3 half), K=0,1 unused.
    const int  mo = m & 7;               // 0..3 valid, 8..11 -> 0..3 duplicate
    const bool va = (mo < NO);
    v2f A2[3];
    #pragma unroll
    for (int c = 0; c < 2; ++c) {
        int k0 = 4 * c + (lo ? 0 : 2);
        A2[c].x = (va) ? W2[mo * NH + k0]     : 0.0f;
        A2[c].y = (va) ? W2[mo * NH + k0 + 1] : 0.0f;
    }
    A2[2].x = (va && !lo) ? W2[mo * NH + 8] : 0.0f;   // K=2 -> h8
    A2[2].y = (va && !lo) ? W2[mo * NH + 9] : 0.0f;   // K=3 -> h9

    // ---- Biases pre-broadcast into C layout (seed the accumulators)
    v8f Cb1, Cb2;
    #pragma unroll
    for (int r = 0; r < 8; ++r) {
        int h = lo ? r : ((r < 2) ? r + 8 : r);   // high half: h8,h9,h2..h7
        Cb1[r] = b1[h];
        Cb2[r] = (r < NO) ? b2[r] : 0.0f;
    }

    // ---- State carried across the T recurrence.
    // Spike doubles as next step's reset: reset_t = (mem_{t-1} > THR) = spk_{t-1}.
    float m1[8], s1[8];
    #pragma unroll
    for (int r = 0; r < 8; ++r) { m1[r] = 0.0f; s1[r] = 0.0f; }
    float m2[4] = {0.0f, 0.0f, 0.0f, 0.0f};
    float s2[4] = {0.0f, 0.0f, 0.0f, 0.0f};

    // ---- Per-lane streaming pointers: low lanes write spike arrays, high
    // lanes write membrane arrays -> every record store is full-wave.
    float* ps01 = out + (b0 + m) * NH + hbase;              // spk1 h0,h1|h8,h9
    float* pm01 = ps01 + o_mem1;                            // mem1 h0,h1|h8,h9
    float* p27  = out + (lo ? 0 : o_mem1) + (b0 + m) * NH;  // h2..h7: spk1|mem1
    float* pl2  = out + (lo ? o_spk2 : o_mem2) + (b0 + m) * NO;

    // ---- Software-pipelined x load (full-wave b64; halves read same address)
    long xoff = (b0 + m) * 2;
    v2f  xv   = { x[xoff], x[xoff + 1] };

    for (int t = 0; t < T; ++t) {
        // Issue next timestep's load early (clamped on the last iteration)
        const long xoff_n = xoff + ((t + 1 < T) ? 2 * B : 0);
        v2f xnext = { x[xoff_n], x[xoff_n + 1] };

        // B operand, layer 1: K=0,1 on low lanes, K=2,3 pad = 0
        v2f Bx = { lo ? xv.x : 0.0f, lo ? xv.y : 0.0f };

        // syn1^T[16x16] = W1dup[16x4] x x^T[4x16] + b1
        v8f syn1 = __builtin_amdgcn_wmma_f32_16x16x4_f32(
            false, A1, false, Bx, (short)0, Cb1, false, false);

        // LIF layer 1: mem = 0.5*mem + syn - spk_prev ; spk = (mem > 1)
        #pragma unroll
        for (int r = 0; r < 8; ++r) {
            m1[r] = 0.5f * m1[r] + syn1[r] - s1[r];
            s1[r] = (m1[r] > 1.0f) ? 1.0f : 0.0f;
        }

        // ---- Record spk1 / mem1: 5 full-wave b64 NT stores, no divergence
        v2f v01s = { s1[0], s1[1] };
        v2f v01m = { m1[0], m1[1] };
        __builtin_nontemporal_store(v01s, (v2f*)ps01);
        __builtin_nontemporal_store(v01m, (v2f*)pm01);
        #pragma unroll
        for (int r = 2; r < 8; r += 2) {   // low: spk1 h_r ; high: mem1 h_r (dup)
            v2f q = { lo ? s1[r] : m1[r], lo ? s1[r + 1] : m1[r + 1] };
            __builtin_nontemporal_store(q, (v2f*)(p27 + r));
        }

        // ---- Layer-2 B chunks: pure per-lane selects between duplicates
        v2f B20 = { lo ? s1[0] : s1[2], lo ? s1[1] : s1[3] };  // h0,h1 | h2,h3
        v2f B21 = { lo ? s1[4] : s1[6], lo ? s1[5] : s1[7] };  // h4,h5 | h6,h7
        v2f B22 = { s1[0], s1[1] };                            // (x)   | h8,h9

        // syn2^T = W2dup x spk1^T + b2 (3 chained K=4 WMMAs)
        v8f c2 = __builtin_amdgcn_wmma_f32_16x16x4_f32(
            false, A2[0], false, B20, (short)0, Cb2, false, false);
        c2 = __builtin_amdgcn_wmma_f32_16x16x4_f32(
            false, A2[1], false, B21, (short)0, c2, false, false);
        c2 = __builtin_amdgcn_wmma_f32_16x16x4_f32(
            false, A2[2], false, B22, (short)0, c2, false, false);

        // ---- LIF layer 2 + record: one full-wave b128 NT store
        v4f q2;
        #pragma unroll
        for (int o = 0; o < 4; ++o) {
            m2[o] = 0.5f * m2[o] + c2[o] - s2[o];
            s2[o] = (m2[o] > 1.0f) ? 1.0f : 0.0f;
            q2[o] = lo ? s2[o] : m2[o];    // low: spk2 ; high: mem2 (dup)
        }
        __builtin_nontemporal_store(q2, (v4f*)pl2);

        // advance streams
        xv    = xnext;
        xoff  = xoff_n;
        ps01 += B * NH;
        pm01 += B * NH;
        p27  += B * NH;
        pl2  += B * NO;
    }
}

extern "C" void kernel_launch(void* const* d_in, const int* in_sizes, int n_in,
                              void* d_out, int out_size, void* d_ws, size_t ws_size,
                              hipStream_t stream) {
    (void)in_sizes; (void)n_in; (void)out_size; (void)d_ws; (void)ws_size;
    const float* x  = (const float*)d_in[0];
    const float* W1 = (const float*)d_in[1];
    const float* b1 = (const float*)d_in[2];
    const float* W2 = (const float*)d_in[3];
    const float* b2 = (const float*)d_in[4];
    float* out = (float*)d_out;

    // 16 batch elements per wave, 8 waves per block -> 128 per block; B=65536
    const int blocks = 65536 / 128;   // 512
    snn_lif_wmma_kernel<<<blocks, 256, 0, stream>>>(x, W1, b1, W2, b2, out);
}